// SparsemaxAttention_26388279067319
// MI455X (gfx1250) — compile-verified
//
#include <hip/hip_runtime.h>

#define HID    768
#define SEQ    1024
#define BATCH  4
#define HEADS  12
#define HDIM   64
#define BN     (BATCH * SEQ)            // 4096 rows for projections
#define QKV_PER (SEQ * HDIM)            // 65536 floats per (b,h)
#define QKV_TOT ((size_t)BATCH * HEADS * QKV_PER)  // 3,145,728 floats
#define KCH    16
#define NCH    (HID / KCH)              // 48 K-chunks for projections

typedef __attribute__((ext_vector_type(2))) float v2f;
typedef __attribute__((ext_vector_type(8))) float v8f;

__device__ __forceinline__ v8f wmma_f32_4(v2f a, v2f b, v8f c) {
  // 8 args: (neg_a, A, neg_b, B, c_mod, C, reuse_a, reuse_b)
  return __builtin_amdgcn_wmma_f32_16x16x4_f32(false, a, false, b, (short)0, c,
                                               false, false);
}

// ---------------------------------------------------------------------------
// Projection GEMM: out = X[BN x HID] * W[HID x HID] + bias
// qkv_mode=1: scatter to [B, H, N, D]; qkv_mode=0: row-major [BN, HID]
// 128 threads = 4 waves; wave tile 32(M) x 64(N); block tile 128 x 64.
// B tile (16x64) is staged in LDS via async copies, double buffered, and
// shared by all 4 waves (4x reduction in global B traffic).
// ---------------------------------------------------------------------------
__global__ __launch_bounds__(128)
void proj_gemm_kernel(const float* __restrict__ X, const float* __restrict__ W,
                      const float* __restrict__ bias, float* __restrict__ out,
                      int qkv_mode) {
  __shared__ float btile[2][KCH * 64];   // 2 x 4 KB

  const int tid  = threadIdx.x;
  const int lane = tid & 31;
  const int wave = tid >> 5;
  const int m16  = lane & 15;
  const int kh   = (lane >> 4) << 1;     // 0 or 2 (K-pair per A/B layout)
  const int rowBase = blockIdx.y * 128 + wave * 32;
  const int colBase = blockIdx.x * 64;

  // async-copy mapping: chunk = 16 rows x 64 cols = 256 x 16B; 2 pieces/thread
  const int r0 = tid >> 4,        c0 = (tid & 15) << 2;
  const int r1 = (tid + 128) >> 4, c1 = (tid & 15) << 2;
  const float* wbase = W + colBase;

  auto issue_chunk = [&](int ch, int bsel) {
    const float* g0 = wbase + (size_t)(ch * KCH + r0) * HID + c0;
    const float* g1 = wbase + (size_t)(ch * KCH + r1) * HID + c1;
    unsigned l0 = (unsigned)(uintptr_t)&btile[bsel][r0 * 64 + c0];
    unsigned l1 = (unsigned)(uintptr_t)&btile[bsel][r1 * 64 + c1];
    asm volatile("global_load_async_to_lds_b128 %0, %1, off"
                 :: "v"(l0), "v"((unsigned long long)(uintptr_t)g0) : "memory");
    asm volatile("global_load_async_to_lds_b128 %0, %1, off"
                 :: "v"(l1), "v"((unsigned long long)(uintptr_t)g1) : "memory");
  };

  v8f acc[2][4];
#pragma unroll
  for (int s = 0; s < 2; ++s)
#pragma unroll
    for (int t = 0; t < 4; ++t)
#pragma unroll
      for (int j = 0; j < 8; ++j) acc[s][t][j] = 0.0f;

  const float* arow0 = X + (size_t)(rowBase + m16) * HID;
  const float* arow1 = arow0 + (size_t)16 * HID;

  int bsel = 0;
  issue_chunk(0, 0);
  for (int ch = 0; ch < NCH; ++ch) {
    __syncthreads();                       // everyone done reading buf bsel^1
    if (ch + 1 < NCH) {
      issue_chunk(ch + 1, bsel ^ 1);
      asm volatile("s_wait_asynccnt 0x2" ::: "memory");  // chunk ch landed
    } else {
      asm volatile("s_wait_asynccnt 0x0" ::: "memory");
    }
    __syncthreads();                       // all waves see chunk ch in LDS

    const float* bt = &btile[bsel][0];
    const int kk = ch * KCH;
#pragma unroll
    for (int ks = 0; ks < KCH; ks += 4) {
      v2f a0, a1;
      a0.x = arow0[kk + ks + kh];
      a0.y = arow0[kk + ks + kh + 1];
      a1.x = arow1[kk + ks + kh];
      a1.y = arow1[kk + ks + kh + 1];
      const float* brow = bt + (ks + kh) * 64;
#pragma unroll
      for (int t = 0; t < 4; ++t) {
        v2f b;
        b.x = brow[t * 16 + m16];
        b.y = brow[64 + t * 16 + m16];
        acc[0][t] = wmma_f32_4(a0, b, acc[0][t]);
        acc[1][t] = wmma_f32_4(a1, b, acc[1][t]);
      }
    }
    bsel ^= 1;
  }

#pragma unroll
  for (int t = 0; t < 4; ++t) {
    const int n  = colBase + t * 16 + m16;
    const float bn = bias[n];
#pragma unroll
    for (int s = 0; s < 2; ++s) {
#pragma unroll
      for (int j = 0; j < 8; ++j) {
        const int m = rowBase + s * 16 + ((lane >> 4) << 3) + j;  // C/D layout
        const float val = acc[s][t][j] + bn;
        if (qkv_mode) {
          const int bb = m >> 10, nn = m & (SEQ - 1);
          const int hh = n >> 6,  dd = n & (HDIM - 1);
          out[((size_t)(bb * HEADS + hh) * SEQ + nn) * HDIM + dd] = val;
        } else {
          out[(size_t)m * HID + n] = val;
        }
      }
    }
  }
}

// ---------------------------------------------------------------------------
// Scores: per (b,h): S[n][m] = (q[n,:] . k[m,:]) / 8  -> weights buffer
// wave tile 32(M) x 64(N); per-lane contiguous float2 loads for both q and kT.
// ---------------------------------------------------------------------------
__global__ __launch_bounds__(128)
void scores_kernel(const float* __restrict__ Qb, const float* __restrict__ Kb,
                   float* __restrict__ Wout) {
  const int z = blockIdx.z;                           // 0..47
  const float* Q = Qb + (size_t)z * QKV_PER;
  const float* K = Kb + (size_t)z * QKV_PER;
  float* O = Wout + (size_t)z * SEQ * SEQ;

  const int lane = threadIdx.x & 31;
  const int wave = threadIdx.x >> 5;
  const int m16  = lane & 15;
  const int kh   = (lane >> 4) << 1;
  const int rowBase = blockIdx.y * 128 + wave * 32;
  const int colBase = blockIdx.x * 64;

  v8f acc[2][4];
#pragma unroll
  for (int s = 0; s < 2; ++s)
#pragma unroll
    for (int t = 0; t < 4; ++t)
#pragma unroll
      for (int j = 0; j < 8; ++j) acc[s][t][j] = 0.0f;

  const float* arow0 = Q + (size_t)(rowBase + m16) * HDIM;
  const float* arow1 = arow0 + (size_t)16 * HDIM;
#pragma unroll 4
  for (int k = 0; k < HDIM; k += 4) {
    v2f a0, a1;
    a0.x = arow0[k + kh];
    a0.y = arow0[k + kh + 1];
    a1.x = arow1[k + kh];
    a1.y = arow1[k + kh + 1];
#pragma unroll
    for (int t = 0; t < 4; ++t) {
      const int n = colBase + t * 16 + m16;
      const float* brow = K + (size_t)n * HDIM;       // kT: contiguous per lane
      v2f b;
      b.x = brow[k + kh];
      b.y = brow[k + kh + 1];
      acc[0][t] = wmma_f32_4(a0, b, acc[0][t]);
      acc[1][t] = wmma_f32_4(a1, b, acc[1][t]);
    }
  }

#pragma unroll
  for (int t = 0; t < 4; ++t) {
    const int n = colBase + t * 16 + m16;
#pragma unroll
    for (int s = 0; s < 2; ++s)
#pragma unroll
      for (int j = 0; j < 8; ++j) {
        const int m = rowBase + s * 16 + ((lane >> 4) << 3) + j;
        O[(size_t)m * SEQ + n] = acc[s][t][j] * 0.125f;  // 1/sqrt(64)
      }
  }
}

// ---------------------------------------------------------------------------
// Sparsemax in place: one 256-thread block per row of 1024 scores.
// Bisection on f(tau)=sum relu(z-tau)-1 (monotone), then exact refinement.
// ---------------------------------------------------------------------------
__global__ __launch_bounds__(256)
void sparsemax_kernel(float* __restrict__ w) {
  const int row = blockIdx.x;
  float* p = w + (size_t)row * SEQ;
  const int t   = threadIdx.x;
  const int lid = t & 31;
  const int wid = t >> 5;

  float4 zv = reinterpret_cast<const float4*>(p)[t];

  __shared__ float red[8];
  __shared__ float red2[8];

  // row max
  float m = fmaxf(fmaxf(zv.x, zv.y), fmaxf(zv.z, zv.w));
#pragma unroll
  for (int off = 16; off > 0; off >>= 1)
    m = fmaxf(m, __shfl_xor(m, off, 32));
  if (lid == 0) red[wid] = m;
  __syncthreads();
  m = red[0];
#pragma unroll
  for (int i = 1; i < 8; ++i) m = fmaxf(m, red[i]);

  // tau in [max-1, max)
  float lo = m - 1.0f, hi = m;
  for (int it = 0; it < 24; ++it) {
    const float mid = 0.5f * (lo + hi);
    float s = fmaxf(zv.x - mid, 0.0f) + fmaxf(zv.y - mid, 0.0f) +
              fmaxf(zv.z - mid, 0.0f) + fmaxf(zv.w - mid, 0.0f);
#pragma unroll
    for (int off = 16; off > 0; off >>= 1) s += __shfl_xor(s, off, 32);
    __syncthreads();
    if (lid == 0) red[wid] = s;
    __syncthreads();
    float tot = red[0];
#pragma unroll
    for (int i = 1; i < 8; ++i) tot += red[i];
    if (tot > 1.0f) lo = mid; else hi = mid;     // uniform branch per block
  }

  // exact refinement on identified support
  const float tau0 = 0.5f * (lo + hi);
  float s = 0.0f, c = 0.0f;
  if (zv.x > tau0) { s += zv.x; c += 1.0f; }
  if (zv.y > tau0) { s += zv.y; c += 1.0f; }
  if (zv.z > tau0) { s += zv.z; c += 1.0f; }
  if (zv.w > tau0) { s += zv.w; c += 1.0f; }
#pragma unroll
  for (int off = 16; off > 0; off >>= 1) {
    s += __shfl_xor(s, off, 32);
    c += __shfl_xor(c, off, 32);
  }
  __syncthreads();
  if (lid == 0) { red[wid] = s; red2[wid] = c; }
  __syncthreads();
  float S = red[0], Kc = red2[0];
#pragma unroll
  for (int i = 1; i < 8; ++i) { S += red[i]; Kc += red2[i]; }

  const float tau = (S - 1.0f) / Kc;
  zv.x = fmaxf(zv.x - tau, 0.0f);
  zv.y = fmaxf(zv.y - tau, 0.0f);
  zv.z = fmaxf(zv.z - tau, 0.0f);
  zv.w = fmaxf(zv.w - tau, 0.0f);
  reinterpret_cast<float4*>(p)[t] = zv;
}

// ---------------------------------------------------------------------------
// Context: per (b,h): ctx[n][d] = sum_m w[n][m] * v[m][d], scatter to [B,N,H*D]
// wave tile 32(M) x 64(N).
// ---------------------------------------------------------------------------
__global__ __launch_bounds__(128)
void ctx_kernel(const float* __restrict__ Wgt, const float* __restrict__ Vb,
                float* __restrict__ ctx) {
  const int z = blockIdx.z;
  const float* Wz = Wgt + (size_t)z * SEQ * SEQ;
  const float* V  = Vb + (size_t)z * QKV_PER;
  const int bb = z / HEADS, hh = z % HEADS;

  const int lane = threadIdx.x & 31;
  const int wave = threadIdx.x >> 5;
  const int m16  = lane & 15;
  const int kh   = (lane >> 4) << 1;
  const int rowBase = blockIdx.y * 128 + wave * 32;

  v8f acc[2][4];
#pragma unroll
  for (int s = 0; s < 2; ++s)
#pragma unroll
    for (int t = 0; t < 4; ++t)
#pragma unroll
      for (int j = 0; j < 8; ++j) acc[s][t][j] = 0.0f;

  const float* arow0 = Wz + (size_t)(rowBase + m16) * SEQ;
  const float* arow1 = arow0 + (size_t)16 * SEQ;
#pragma unroll 4
  for (int k = 0; k < SEQ; k += 4) {
    v2f a0, a1;
    a0.x = arow0[k + kh];
    a0.y = arow0[k + kh + 1];
    a1.x = arow1[k + kh];
    a1.y = arow1[k + kh + 1];
    const float* b0 = V + (size_t)(k + kh) * HDIM;
#pragma unroll
    for (int t = 0; t < 4; ++t) {
      v2f b;
      b.x = b0[t * 16 + m16];
      b.y = b0[HDIM + t * 16 + m16];
      acc[0][t] = wmma_f32_4(a0, b, acc[0][t]);
      acc[1][t] = wmma_f32_4(a1, b, acc[1][t]);
    }
  }

#pragma unroll
  for (int t = 0; t < 4; ++t) {
    const int d = t * 16 + m16;                 // 0..63
#pragma unroll
    for (int s = 0; s < 2; ++s)
#pragma unroll
      for (int j = 0; j < 8; ++j) {
        const int m = rowBase + s * 16 + ((lane >> 4) << 3) + j;
        ctx[((size_t)bb * SEQ + m) * HID + hh * HDIM + d] = acc[s][t][j];
      }
  }
}

// ---------------------------------------------------------------------------
extern "C" void kernel_launch(void* const* d_in, const int* in_sizes, int n_in,
                              void* d_out, int out_size, void* d_ws,
                              size_t ws_size, hipStream_t stream) {
  const float* x  = (const float*)d_in[0];
  const float* Wq = (const float*)d_in[1];
  const float* bq = (const float*)d_in[2];
  const float* Wk = (const float*)d_in[3];
  const float* bk = (const float*)d_in[4];
  const float* Wv = (const float*)d_in[5];
  const float* bv = (const float*)d_in[6];
  const float* Wo = (const float*)d_in[7];
  const float* bo = (const float*)d_in[8];

  float* out  = (float*)d_out;
  float* attw = out + (size_t)BN * HID;   // attn_weights region of d_out

  float* qbuf = (float*)d_ws;
  float* kbuf = qbuf + QKV_TOT;
  float* vbuf = kbuf + QKV_TOT;
  float* ctxb = vbuf + QKV_TOT;

  const dim3 blk(128);
  const dim3 gProj(HID / 64, BN / 128);                // (12, 32)
  proj_gemm_kernel<<<gProj, blk, 0, stream>>>(x, Wq, bq, qbuf, 1);
  proj_gemm_kernel<<<gProj, blk, 0, stream>>>(x, Wk, bk, kbuf, 1);
  proj_gemm_kernel<<<gProj, blk, 0, stream>>>(x, Wv, bv, vbuf, 1);

  const dim3 gScore(SEQ / 64, SEQ / 128, BATCH * HEADS);  // (16, 8, 48)
  scores_kernel<<<gScore, blk, 0, stream>>>(qbuf, kbuf, attw);

  sparsemax_kernel<<<dim3(BATCH * HEADS * SEQ), dim3(256), 0, stream>>>(attw);

  const dim3 gCtx(1, SEQ / 128, BATCH * HEADS);           // (1, 8, 48)
  ctx_kernel<<<gCtx, blk, 0, stream>>>(attw, vbuf, ctxb);

  proj_gemm_kernel<<<gProj, blk, 0, stream>>>(ctxb, Wo, bo, out, 0);
}